// canny_edge_5437428596810
// MI455X (gfx1250) — compile-verified
//
#include <hip/hip_runtime.h>
#include <stdint.h>

#define H 512
#define W 512
#define PLANE (H * W)   // 262144
#define TS 64

typedef float v4f __attribute__((ext_vector_type(4)));   // native clang vector (NT-builtin friendly)

// ---- CDNA5 async global->LDS staging helpers (gfx1250, compile-proven round 1) ----
__device__ __forceinline__ void async_ld_b32(uint32_t lds_addr, const void* gptr) {
  unsigned long long ga = (unsigned long long)(uintptr_t)gptr;
  // GLOBAL_LOAD_ASYNC_TO_LDS_B32: VDST = per-lane LDS byte address, VADDR = 64-bit global addr (GV mode)
  asm volatile("global_load_async_to_lds_b32 %0, %1, off" :: "v"(lds_addr), "v"(ga) : "memory");
}
__device__ __forceinline__ void wait_async0() {
  asm volatile("s_wait_asynccnt 0" ::: "memory");
}
__device__ __forceinline__ uint32_t lds_off(const void* p) {
  // generic LDS pointer: low 32 bits are the wave-relative LDS byte address
  return (uint32_t)(uintptr_t)p;
}

// ------------------- K1: channel shuffle (x5) -------------------
// RT load (x is re-read by k_canny -> keep in L2), NT store (x5 never re-read).
__global__ __launch_bounds__(256) void k_shuffle(const float* __restrict__ x,
                                                 const int* __restrict__ perm,
                                                 float* __restrict__ x5) {
  const int plane = blockIdx.y;          // b*3 + c
  const int b = plane / 3;
  const int p = perm[plane];             // source channel
  const v4f* __restrict__ src = (const v4f*)(x + ((size_t)(b * 3 + p)) * PLANE);
  v4f* __restrict__ dst = (v4f*)(x5 + (size_t)plane * PLANE);
  const int i = blockIdx.x * blockDim.x + threadIdx.x;   // 65536 float4 per plane
  __builtin_nontemporal_store(src[i], &dst[i]);
}

// ------------- K2: gray -> blur -> sobel -> mag -> NMS -------------
// tile 64x64, gray halo 4 (blur 2 + sobel 1 + NMS 1)
__global__ __launch_bounds__(256) void k_canny(const float* __restrict__ x,
                                               float* __restrict__ out_mag) {
  __shared__ float sm[3 * 5256];         // 3 buffers of 72x73 (padded stride), ~62 KB
  float* p0 = sm;                        // ch0 -> gray -> mag
  float* p1 = sm + 5256;                 // ch1 -> hblur
  float* p2 = sm + 2 * 5256;             // ch2 -> blurred

  const int b   = blockIdx.z;
  const int ty0 = blockIdx.y * TS;
  const int tx0 = blockIdx.x * TS;
  const int tid = threadIdx.x;
  const float* __restrict__ xb = x + (size_t)b * 3 * PLANE;

  // stage 0: async-stage 3 channel tiles (72x72), reflect indexing at image borders
  for (int e = tid; e < 72 * 72; e += 256) {
    const int ly = e / 72, lx = e - ly * 72;
    int gy = ty0 - 4 + ly; gy = (gy < 0) ? -gy : ((gy > 511) ? 1022 - gy : gy);
    int gx = tx0 - 4 + lx; gx = (gx < 0) ? -gx : ((gx > 511) ? 1022 - gx : gx);
    const float* g0 = xb + (size_t)gy * W + gx;
    const int li = ly * 73 + lx;
    async_ld_b32(lds_off(&p0[li]), g0);
    async_ld_b32(lds_off(&p1[li]), g0 + PLANE);
    async_ld_b32(lds_off(&p2[li]), g0 + 2 * PLANE);
  }
  wait_async0();
  __syncthreads();

  // stage 1: gray = 0.114*B + 0.587*G + 0.299*R  (in-place over p0)
  for (int e = tid; e < 72 * 72; e += 256) {
    const int ly = e / 72, lx = e - ly * 72;
    const int i = ly * 73 + lx;
    p0[i] = 0.114f * p0[i] + 0.587f * p1[i] + 0.299f * p2[i];
  }
  __syncthreads();

  const float G0 = 0.054488684f, G1 = 0.244201342f, G2 = 0.402619947f;

  // stage 2: horizontal gaussian: hb (72 rows x 68 cols, stride 69) into p1
  for (int e = tid; e < 72 * 68; e += 256) {
    const int ly = e / 68, lx = e - ly * 68;
    const float* r = &p0[ly * 73 + lx];
    p1[ly * 69 + lx] = G0 * (r[0] + r[4]) + G1 * (r[1] + r[3]) + G2 * r[2];
  }
  __syncthreads();

  // stage 3: vertical gaussian: blurred (68x68, stride 69) into p2
  for (int e = tid; e < 68 * 68; e += 256) {
    const int ly = e / 68, lx = e - ly * 68;
    const float* c = &p1[ly * 69 + lx];
    p2[ly * 69 + lx] = G0 * (c[0] + c[4 * 69]) + G1 * (c[69] + c[3 * 69]) + G2 * c[2 * 69];
  }
  __syncthreads();

  // blurred sampled with replicate (clamp) at image borders; local origin = (ty0-2, tx0-2)
  auto BL = [&](int gY, int gX) -> float {
    const int cy = gY < 0 ? 0 : (gY > 511 ? 511 : gY);
    const int cx = gX < 0 ? 0 : (gX > 511 ? 511 : gX);
    return p2[(cy - (ty0 - 2)) * 69 + (cx - (tx0 - 2))];
  };
  auto SOB = [&](int Y, int X, float& sgx, float& sgy) {
    const float tl = BL(Y - 1, X - 1), tc = BL(Y - 1, X), tr = BL(Y - 1, X + 1);
    const float ml = BL(Y, X - 1),                         mr = BL(Y, X + 1);
    const float bl = BL(Y + 1, X - 1), bc = BL(Y + 1, X), br = BL(Y + 1, X + 1);
    sgx = (tr - tl) + 2.f * (mr - ml) + (br - bl);
    sgy = (bl - tl) + 2.f * (bc - tc) + (br - tr);
  };

  // stage 4: mag on 66x66 (stride 67) into p0 (gray no longer needed)
  for (int e = tid; e < 66 * 66; e += 256) {
    const int i = e / 66, j = e - i * 66;
    const int Y = ty0 - 1 + i, X = tx0 - 1 + j;
    float sgx, sgy;
    SOB(Y, X, sgx, sgy);
    p0[i * 67 + j] = sqrtf(sgx * sgx + sgy * sgy + 1e-6f);
  }
  __syncthreads();

  // stage 5: NMS on 64x64, write mag output (RT store: k_dilate re-reads it from L2)
  float* __restrict__ dst = out_mag + (size_t)b * PLANE;
  for (int e = tid; e < 64 * 64; e += 256) {
    const int y = e >> 6, xx = e & 63;
    const int Y = ty0 + y, X = tx0 + xx;
    float sgx, sgy;
    SOB(Y, X, sgx, sgy);
    const float ax = fabsf(sgx), ay = fabsf(sgy);
    int dy, dx;
    if (ay <= 0.41421356f * ax)      { dy = 0; dx = 1; }               // horizontal bin
    else if (ay >= 2.41421356f * ax) { dy = 1; dx = 0; }               // vertical bin
    else { dy = 1; dx = ((sgx > 0.f) == (sgy > 0.f)) ? 1 : -1; }       // diagonal bins
    const float m = p0[(y + 1) * 67 + (xx + 1)];
    float n1 = 0.f, n2 = 0.f;
    if ((unsigned)(Y + dy) < 512u && (unsigned)(X + dx) < 512u)
      n1 = p0[(y + 1 + dy) * 67 + (xx + 1 + dx)];
    if ((unsigned)(Y - dy) < 512u && (unsigned)(X - dx) < 512u)
      n2 = p0[(y + 1 - dy) * 67 + (xx + 1 - dx)];
    const float r = ((m - n1) > 0.f && (m - n2) > 0.f) ? m : 0.f;
    dst[(size_t)Y * W + X] = r;
  }
}

// ------------- K3: dilations 3/5/7/9 as a maxpool3 cascade -------------
__global__ __launch_bounds__(256) void k_dilate(const float* __restrict__ mag,
                                                float* __restrict__ out) {
  __shared__ float A[5256];   // 72x73
  __shared__ float B[5256];
  const int b   = blockIdx.z;
  const int ty0 = blockIdx.y * TS;
  const int tx0 = blockIdx.x * TS;
  const int tid = threadIdx.x;
  const float* __restrict__ mp = mag + (size_t)b * PLANE;

  // stage mag tile (halo 4) into LDS: async loads in-image, 0-fill outside
  for (int e = tid; e < 72 * 72; e += 256) {
    const int ly = e / 72, lx = e - ly * 72;
    const int gy = ty0 - 4 + ly, gx = tx0 - 4 + lx;
    const int li = ly * 73 + lx;
    if ((unsigned)gy < 512u && (unsigned)gx < 512u) {
      async_ld_b32(lds_off(&A[li]), mp + (size_t)gy * W + gx);
    } else {
      A[li] = 0.f;   // mag >= 0, so 0-fill == clipped SAME window
    }
  }
  wait_async0();
  __syncthreads();

  for (int s = 0; s < 4; ++s) {
    const int R = 72 - 2 * s;           // current region size in A
    // horizontal max3: B[y][x] = max(A[y][x..x+2]), region R x (R-2)
    for (int e = tid; e < R * (R - 2); e += 256) {
      const int y = e / (R - 2), xx = e - y * (R - 2);
      const float* r = &A[y * 73 + xx];
      B[y * 73 + xx] = fmaxf(fmaxf(r[0], r[1]), r[2]);
    }
    __syncthreads();
    // vertical max3: A[y][x] = max(B[y..y+2][x]), region (R-2) x (R-2)
    for (int e = tid; e < (R - 2) * (R - 2); e += 256) {
      const int y = e / (R - 2), xx = e - y * (R - 2);
      const float* c = &B[y * 73 + xx];
      A[y * 73 + xx] = fmaxf(fmaxf(c[0], c[73]), c[2 * 73]);
    }
    __syncthreads();
    // write d_{3+2s} with NT stores (never re-read); output origin offset in A is h = 3 - s
    const int h = 3 - s;
    float* __restrict__ dst = out + (size_t)(16 * (s + 1) + b) * PLANE;
    for (int e = tid; e < 64 * 64; e += 256) {
      const int y = e >> 6, xx = e & 63;
      __builtin_nontemporal_store(A[(y + h) * 73 + (xx + h)],
                                  &dst[(size_t)(ty0 + y) * W + (tx0 + xx)]);
    }
    __syncthreads();
  }
}

// ------------------------- launch -------------------------
extern "C" void kernel_launch(void* const* d_in, const int* in_sizes, int n_in,
                              void* d_out, int out_size, void* d_ws, size_t ws_size,
                              hipStream_t stream) {
  (void)in_sizes; (void)n_in; (void)out_size; (void)d_ws; (void)ws_size;
  const float* x    = (const float*)d_in[0];
  const int*   perm = (const int*)d_in[1];
  // d_in[2..5] are the fixed dilation sizes 3,5,7,9 (baked into the cascade)
  float* out = (float*)d_out;

  // output layout: [mag | d3 | d5 | d7 | d9 | x5], each segment 16*PLANE (x5 = 48*PLANE)
  float* x5 = out + (size_t)80 * PLANE;

  k_shuffle<<<dim3(PLANE / 4 / 256, 48), 256, 0, stream>>>(x, perm, x5);
  k_canny  <<<dim3(8, 8, 16),            256, 0, stream>>>(x, out);
  k_dilate <<<dim3(8, 8, 16),            256, 0, stream>>>(out, out);
}